// AttnBlock_80771154968680
// MI455X (gfx1250) — compile-verified
//
#include <hip/hip_runtime.h>
#include <hip/hip_bf16.h>
#include <math.h>

typedef float v2f __attribute__((ext_vector_type(2)));
typedef float v8f __attribute__((ext_vector_type(8)));

#define C_TOT 512
#define HW    1024
#define NB    4
#define GROUPS 32
#define CPG   16        // channels per group
#define NHEAD 8
#define CH    64        // channels per head

// ---------------------------------------------------------------------------
// GroupNorm: one block per (group, batch). Group = contiguous 16*1024 floats.
// ---------------------------------------------------------------------------
__global__ __launch_bounds__(256) void gn_kernel(
    const float* __restrict__ X, const float* __restrict__ scale,
    const float* __restrict__ bias, float* __restrict__ Y)
{
    const int g = blockIdx.x;
    const int b = blockIdx.y;
    const size_t base = ((size_t)b * C_TOT + (size_t)g * CPG) * HW;
    const int N = CPG * HW;   // 16384

    float s = 0.f, s2 = 0.f;
    for (int i = threadIdx.x; i < N; i += 256) {
        float v = X[base + i];
        s += v; s2 += v * v;
    }
    __shared__ float red[256];
    __shared__ float red2[256];
    red[threadIdx.x] = s; red2[threadIdx.x] = s2;
    __syncthreads();
    for (int off = 128; off > 0; off >>= 1) {
        if (threadIdx.x < off) {
            red[threadIdx.x]  += red[threadIdx.x + off];
            red2[threadIdx.x] += red2[threadIdx.x + off];
        }
        __syncthreads();
    }
    const float inv_n = 1.0f / (float)N;
    const float mean = red[0] * inv_n;
    const float var  = red2[0] * inv_n - mean * mean;
    const float rstd = rsqrtf(var + 1e-6f);

    for (int i = threadIdx.x; i < N; i += 256) {
        int c = g * CPG + (i >> 10);          // i / HW
        float v = (X[base + i] - mean) * rstd;
        Y[base + i] = v * scale[c] + bias[c];
    }
}

// ---------------------------------------------------------------------------
// conv1x1 as GEMM: Y[b][o][n] = (sum_c W[o][c] * X[b][c][n] + bias[o]) * scale
//                  (+ residual). One wave per 16x16 output tile.
// A = W tile (16 o x 4 c), B = X tile (4 c x 16 n), f32 WMMA, K = 512.
// ---------------------------------------------------------------------------
__global__ __launch_bounds__(32) void conv1x1_wmma(
    const float* __restrict__ W, const float* __restrict__ X,
    const float* __restrict__ bias, const float* __restrict__ res,
    float* __restrict__ Y, float scale)
{
    const int lane = threadIdx.x;
    const int m = lane & 15;      // M index for A rows / N index for B cols
    const int h = lane >> 4;      // lane-half selects K pair (2h, 2h+1)
    const int mb = blockIdx.x * 16;   // output-channel tile
    const int nb = blockIdx.y * 16;   // spatial tile
    const int b  = blockIdx.z;

    const float* Xb   = X + (size_t)b * C_TOT * HW;
    const float* wrow = W + (size_t)(mb + m) * C_TOT;

    v8f acc = {};
#pragma unroll 4
    for (int k0 = 0; k0 < C_TOT; k0 += 4) {
        v2f a = *(const v2f*)(wrow + k0 + 2 * h);          // contiguous pair
        v2f bb;
        bb.x = Xb[(size_t)(k0 + 2 * h)     * HW + nb + m];
        bb.y = Xb[(size_t)(k0 + 2 * h + 1) * HW + nb + m];
        acc = __builtin_amdgcn_wmma_f32_16x16x4_f32(
            false, a, false, bb, (short)0, acc, false, false);
    }

    float* Yb = Y + (size_t)b * C_TOT * HW;
    const float* Rb = res ? (res + (size_t)b * C_TOT * HW) : nullptr;
#pragma unroll
    for (int r = 0; r < 8; ++r) {
        int row = mb + r + 8 * h;                 // output channel
        float val = (acc[r] + bias[row]) * scale;
        size_t idx = (size_t)row * HW + nb + m;   // spatial col = m
        if (Rb) val += Rb[idx];
        Yb[idx] = val;
    }
}

// ---------------------------------------------------------------------------
// Fused flash attention. One wave per (batch*head, 16-query tile).
// Scores never touch HBM: S tile in regs, P via small LDS transpose.
// q already carries the 1/sqrt(ch) scale.
// ---------------------------------------------------------------------------
__global__ __launch_bounds__(32) void attn_wmma(
    const float* __restrict__ Q, const float* __restrict__ K,
    const float* __restrict__ V, float* __restrict__ O)
{
    __shared__ __align__(16) float pl[16 * 18];   // P tile, padded stride 18

    const int lane = threadIdx.x;
    const int m = lane & 15;
    const int h = lane >> 4;
    const int ib = blockIdx.x * 16;               // query tile base
    const int bh = blockIdx.y;                    // 0..31
    const int b = bh >> 3, head = bh & 7;
    const size_t base = ((size_t)b * C_TOT + (size_t)head * CH) * HW;
    const float* qp = Q + base;
    const float* kp = K + base;
    const float* vp = V + base;

    // Preload Q as A fragments: qa[kc] covers channels 4*kc .. 4*kc+3
    v2f qa[16];
#pragma unroll
    for (int kc = 0; kc < 16; ++kc) {
        qa[kc].x = qp[(size_t)(4 * kc + 2 * h)     * HW + ib + m];
        qa[kc].y = qp[(size_t)(4 * kc + 2 * h + 1) * HW + ib + m];
    }

    v8f acc0 = {}, acc1 = {}, acc2 = {}, acc3 = {};
    float mrun[8], lrun[8];
#pragma unroll
    for (int r = 0; r < 8; ++r) { mrun[r] = -3.0e38f; lrun[r] = 0.f; }

    for (int jb = 0; jb < HW; jb += 16) {
        // ---- scores S = q^T k  (16 queries x 16 keys), K-dim = 64 ----
        v8f s = {};
#pragma unroll
        for (int kc = 0; kc < 16; ++kc) {
            v2f kb;
            kb.x = kp[(size_t)(4 * kc + 2 * h)     * HW + jb + m];
            kb.y = kp[(size_t)(4 * kc + 2 * h + 1) * HW + jb + m];
            s = __builtin_amdgcn_wmma_f32_16x16x4_f32(
                false, qa[kc], false, kb, (short)0, s, false, false);
        }
        // ---- online softmax per row (row = r + 8h; cols span lane half) ----
#pragma unroll
        for (int r = 0; r < 8; ++r) {
            float sv = s[r];
            float rmax = sv;
            rmax = fmaxf(rmax, __shfl_xor(rmax, 1, 32));
            rmax = fmaxf(rmax, __shfl_xor(rmax, 2, 32));
            rmax = fmaxf(rmax, __shfl_xor(rmax, 4, 32));
            rmax = fmaxf(rmax, __shfl_xor(rmax, 8, 32));
            float mnew = fmaxf(mrun[r], rmax);
            float corr = __expf(mrun[r] - mnew);
            float p    = __expf(sv - mnew);
            float rsum = p;
            rsum += __shfl_xor(rsum, 1, 32);
            rsum += __shfl_xor(rsum, 2, 32);
            rsum += __shfl_xor(rsum, 4, 32);
            rsum += __shfl_xor(rsum, 8, 32);
            lrun[r] = lrun[r] * corr + rsum;
            mrun[r] = mnew;
            s[r] = p;
            acc0[r] *= corr; acc1[r] *= corr;
            acc2[r] *= corr; acc3[r] *= corr;
        }
        // ---- C-layout P -> LDS -> A-layout fragments ----
#pragma unroll
        for (int r = 0; r < 8; ++r)
            pl[(r + 8 * h) * 18 + m] = s[r];
        __syncthreads();
        // ---- O += P * V^T :  A = P (16 i x 16 j), B[j][c] = V[c][j] ----
#pragma unroll
        for (int jc = 0; jc < 4; ++jc) {
            v2f pa = *(const v2f*)&pl[m * 18 + 4 * jc + 2 * h];
            int j = jb + 4 * jc + 2 * h;
            v2f vb0 = *(const v2f*)(vp + (size_t)(0 * 16 + m) * HW + j);
            v2f vb1 = *(const v2f*)(vp + (size_t)(1 * 16 + m) * HW + j);
            v2f vb2 = *(const v2f*)(vp + (size_t)(2 * 16 + m) * HW + j);
            v2f vb3 = *(const v2f*)(vp + (size_t)(3 * 16 + m) * HW + j);
            acc0 = __builtin_amdgcn_wmma_f32_16x16x4_f32(false, pa, false, vb0, (short)0, acc0, false, false);
            acc1 = __builtin_amdgcn_wmma_f32_16x16x4_f32(false, pa, false, vb1, (short)0, acc1, false, false);
            acc2 = __builtin_amdgcn_wmma_f32_16x16x4_f32(false, pa, false, vb2, (short)0, acc2, false, false);
            acc3 = __builtin_amdgcn_wmma_f32_16x16x4_f32(false, pa, false, vb3, (short)0, acc3, false, false);
        }
        __syncthreads();
    }

    // ---- normalize and write out[c][i] ----
    float* op = O + base;
#pragma unroll
    for (int r = 0; r < 8; ++r) {
        float inv = 1.0f / lrun[r];
        int i = ib + r + 8 * h;                  // query (spatial) index
        op[(size_t)(0 * 16 + m) * HW + i] = acc0[r] * inv;
        op[(size_t)(1 * 16 + m) * HW + i] = acc1[r] * inv;
        op[(size_t)(2 * 16 + m) * HW + i] = acc2[r] * inv;
        op[(size_t)(3 * 16 + m) * HW + i] = acc3[r] * inv;
    }
}

// ---------------------------------------------------------------------------
extern "C" void kernel_launch(void* const* d_in, const int* in_sizes, int n_in,
                              void* d_out, int out_size, void* d_ws, size_t ws_size,
                              hipStream_t stream)
{
    const float* x          = (const float*)d_in[0];
    const float* kv         = (const float*)d_in[1];
    const float* gnq_scale  = (const float*)d_in[2];
    const float* gnq_bias   = (const float*)d_in[3];
    const float* gnkv_scale = (const float*)d_in[4];
    const float* gnkv_bias  = (const float*)d_in[5];
    const float* wq    = (const float*)d_in[6];
    const float* bq    = (const float*)d_in[7];
    const float* wk    = (const float*)d_in[8];
    const float* bk    = (const float*)d_in[9];
    const float* wv    = (const float*)d_in[10];
    const float* bv    = (const float*)d_in[11];
    const float* wproj = (const float*)d_in[12];
    const float* bproj = (const float*)d_in[13];
    float* out = (float*)d_out;

    float* ws = (float*)d_ws;
    const size_t T = (size_t)NB * C_TOT * HW;   // 2M floats = 8 MB
    float* xn  = ws;           // groupnorm(x); reused later as attention output
    float* kvn = ws + T;
    float* q   = ws + 2 * T;
    float* k   = ws + 3 * T;
    float* v   = ws + 4 * T;
    float* o   = xn;           // xn is dead after the q projection

    // 1) GroupNorms
    gn_kernel<<<dim3(GROUPS, NB), 256, 0, stream>>>(x,  gnq_scale,  gnq_bias,  xn);
    gn_kernel<<<dim3(GROUPS, NB), 256, 0, stream>>>(kv, gnkv_scale, gnkv_bias, kvn);

    // 2) q/k/v projections (1/sqrt(64) folded into q)
    dim3 gg(C_TOT / 16, HW / 16, NB);
    conv1x1_wmma<<<gg, 32, 0, stream>>>(wq, xn,  bq, nullptr, q, 0.125f);
    conv1x1_wmma<<<gg, 32, 0, stream>>>(wk, kvn, bk, nullptr, k, 1.0f);
    conv1x1_wmma<<<gg, 32, 0, stream>>>(wv, kvn, bv, nullptr, v, 1.0f);

    // 3) fused flash attention
    attn_wmma<<<dim3(HW / 16, NB * NHEAD), 32, 0, stream>>>(q, k, v, o);

    // 4) output projection + residual
    conv1x1_wmma<<<gg, 32, 0, stream>>>(wproj, o, bproj, x, out, 1.0f);
}